// SPA_59304908423389
// MI455X (gfx1250) — compile-verified
//
#include <hip/hip_runtime.h>

typedef __attribute__((ext_vector_type(2))) float v2f;
typedef __attribute__((ext_vector_type(4))) float v4f;
typedef __attribute__((ext_vector_type(8))) float v8f;
typedef __attribute__((ext_vector_type(4))) unsigned int u32x4;
typedef __attribute__((ext_vector_type(8))) int i32x8;
typedef __attribute__((ext_vector_type(4))) int i32x4;

#define NPIX 32768   // 8*64*64
#define DIMC 384
#define DPLC 96

#if defined(__has_builtin)
#if __has_builtin(__builtin_amdgcn_tensor_load_to_lds)
#define HAVE_TDM 1
#endif
#endif
#ifndef HAVE_TDM
#define HAVE_TDM 0
#endif

#if HAVE_TDM
// ---------------------------------------------------------------------------
// Tensor Data Mover: async 2D f32 tile (rows x rowlen=32, row stride
// stride_elems) -> LDS at lds_byte_off, inserting 1 pad dword every 32
// dwords (LDS row pitch 33 floats = the padded shared arrays below).
// D# layout per CDNA5 ISA.
// ---------------------------------------------------------------------------
__device__ __forceinline__ void tdm_load_tile_f32(const float* gsrc,
                                                  unsigned lds_byte_off,
                                                  int rows, int rowlen,
                                                  int stride_elems)
{
    unsigned long long ga = (unsigned long long)(uintptr_t)gsrc;
    u32x4 g0;
    g0[0] = 1u;                                        // count = 1 (valid D#)
    g0[1] = lds_byte_off;                              // lds_addr (bytes)
    g0[2] = (unsigned)(ga & 0xffffffffu);              // global_addr[31:0]
    g0[3] = (unsigned)((ga >> 32) & 0x1ffffffu)        // global_addr[56:32]
          | (2u << 30);                                // type = 2 ("image")
    i32x8 g1;
    g1[0] = (int)((2u << 16)      // data_size = 2 (4 bytes)
          |       (1u << 20)      // pad_enable
          |       (4u << 22));    // pad_interval code 4 = every 32 dwords
                                  // pad_amount = 0 -> 1 dword
    g1[1] = (int)((unsigned)rowlen << 16);   // tensor_dim0 (low 16)
    g1[2] = (int)((unsigned)rows   << 16);   // tensor_dim1 (low 16)
    g1[3] = (int)((unsigned)rowlen << 16);   // tile_dim0
    g1[4] = (int)rows;                       // tile_dim1 (tile_dim2 = 0)
    g1[5] = stride_elems;                    // tensor_dim0_stride[31:0]
    g1[6] = 0;
    g1[7] = 0;
    i32x4 z4 = {};
#if __clang_major__ >= 23
    i32x8 z8 = {};
    __builtin_amdgcn_tensor_load_to_lds(g0, g1, z4, z4, z8, 0);
#else
    __builtin_amdgcn_tensor_load_to_lds(g0, g1, z4, z4, 0);
#endif
}
#endif // HAVE_TDM

// ---------------------------------------------------------------------------
// Fused WMMA GEMM:  C[M x N] = A[M x 384] * B  (+ epilogue per MODE)
//   MODE 0: B row-major [384 x N]            -> C = y (ldc = 864)
//   MODE 1: B = interact_w^T (W is 384x384)  -> t = (C + bias) * k * v
//   MODE 2: B = proj_w^T                     -> out = C + bias
// Block: 128 threads = 4 waves; tile 128(M) x 32(N), K-step 32, 12 K-tiles.
// Each wave owns a 32x32 output tile = 2x2 WMMA accumulators (1 LDS load
// per WMMA). A (and MODE-0 B) tiles stream in via the Tensor Data Mover
// with 2-deep double buffering: tile i+1's DMA runs behind tile i's
// 32-WMMA compute phase; tensorcnt<=inflight proves tile i is resident
// (TDM ops from one wave complete in order).
// ---------------------------------------------------------------------------
template <int MODE>
__global__ __launch_bounds__(128) void wmma_gemm_kernel(
    const float* __restrict__ A, int lda,
    const float* __restrict__ B,
    const float* __restrict__ bias,
    const float* __restrict__ ykv,   // MODE 1: k/v source (row stride 864)
    float* __restrict__ C, int N)
{
    const int n0   = blockIdx.x * 32;
    const int m0   = blockIdx.y * 128;
    const int tid  = threadIdx.x;
    const int lane = tid & 31;
    const int wv   = tid >> 5;       // wave 0..3 -> 32-row M sub-tile
    const int half = lane >> 4;      // 0: lanes 0-15, 1: lanes 16-31
    const int lr   = lane & 15;

    __shared__ float As[2][128][33]; // double-buffered, pitch 33 (TDM pad)
    __shared__ float Bs[2][32][33];

    v8f acc00 = {}, acc01 = {}, acc10 = {}, acc11 = {};

    constexpr int KT = DIMC / 32;    // 12 K-tiles

    // stage K-tile ki into LDS buffer buf
    auto stage = [&](int ki, int buf) {
        const int k0 = ki * 32;
#if HAVE_TDM
        if (wv == 0) {
            tdm_load_tile_f32(A + (size_t)m0 * lda + k0,
                              (unsigned)(uintptr_t)&As[buf][0][0],
                              128, 32, lda);
            if (MODE == 0) {
                tdm_load_tile_f32(B + (size_t)k0 * N + n0,
                                  (unsigned)(uintptr_t)&Bs[buf][0][0],
                                  32, 32, N);
            }
        }
#else
#pragma unroll
        for (int i = 0; i < 8; ++i) {
            int idx = tid + i * 128;
            int row = idx >> 3;
            int cg  = idx & 7;
            v4f av = *reinterpret_cast<const v4f*>(
                A + (size_t)(m0 + row) * lda + k0 + cg * 4);
#pragma unroll
            for (int j = 0; j < 4; ++j) As[buf][row][cg * 4 + j] = av[j];
        }
        if (MODE == 0) {
#pragma unroll
            for (int i = 0; i < 2; ++i) {
                int idx = tid + i * 128;
                int kr = idx >> 3, cg = idx & 7;
                v4f bv = *reinterpret_cast<const v4f*>(
                    B + (size_t)(k0 + kr) * N + n0 + cg * 4);
#pragma unroll
                for (int j = 0; j < 4; ++j) Bs[buf][kr][cg * 4 + j] = bv[j];
            }
        }
#endif
        if (MODE != 0) {
            // B[k][n] = W[n0+n][k0+k]; transpose-scatter 32x32 block of W
#pragma unroll
            for (int i = 0; i < 2; ++i) {
                int idx = tid + i * 128;
                int nn = idx >> 3, cg = idx & 7;
                v4f bv = *reinterpret_cast<const v4f*>(
                    B + (size_t)(n0 + nn) * DIMC + k0 + cg * 4);
#pragma unroll
                for (int j = 0; j < 4; ++j) Bs[buf][cg * 4 + j][nn] = bv[j];
            }
        }
    };

    stage(0, 0);                     // prologue: tile 0 -> buffer 0

    for (int i = 0; i < KT; ++i) {
        const int cur = i & 1;
        if (i + 1 < KT) {
            stage(i + 1, cur ^ 1);   // tile i+1 streams behind tile i compute
#if HAVE_TDM
            if (wv == 0) {
                if (MODE == 0) __builtin_amdgcn_s_wait_tensorcnt(2);
                else           __builtin_amdgcn_s_wait_tensorcnt(1);
            }
#endif
        } else {
#if HAVE_TDM
            if (wv == 0) __builtin_amdgcn_s_wait_tensorcnt(0);
#endif
        }
        __syncthreads();             // tile i resident & visible to all waves

        // ---- 8 K-substeps x 4 WMMA (2x2 register tile) per wave ----
#pragma unroll
        for (int kk = 0; kk < 8; ++kk) {
            const int kb = kk * 4 + 2 * half;
            v2f a0, a1, b0, b1;
            a0.x = As[cur][wv * 32 + lr][kb];
            a0.y = As[cur][wv * 32 + lr][kb + 1];
            a1.x = As[cur][wv * 32 + 16 + lr][kb];
            a1.y = As[cur][wv * 32 + 16 + lr][kb + 1];
            b0.x = Bs[cur][kb][lr];
            b0.y = Bs[cur][kb + 1][lr];
            b1.x = Bs[cur][kb][16 + lr];
            b1.y = Bs[cur][kb + 1][16 + lr];
            acc00 = __builtin_amdgcn_wmma_f32_16x16x4_f32(
                false, a0, false, b0, (short)0, acc00, false, false);
            acc01 = __builtin_amdgcn_wmma_f32_16x16x4_f32(
                false, a0, false, b1, (short)0, acc01, false, false);
            acc10 = __builtin_amdgcn_wmma_f32_16x16x4_f32(
                false, a1, false, b0, (short)0, acc10, false, false);
            acc11 = __builtin_amdgcn_wmma_f32_16x16x4_f32(
                false, a1, false, b1, (short)0, acc11, false, false);
        }
        __syncthreads();             // free buffer cur for stage(i+2)
    }

    // ---- epilogue: lane lr = col-in-tile, VGPR r = row (half*8 + r) ----
    v8f* accs[2][2] = { { &acc00, &acc01 }, { &acc10, &acc11 } };
#pragma unroll
    for (int mi = 0; mi < 2; ++mi) {
#pragma unroll
        for (int ni = 0; ni < 2; ++ni) {
            v8f a = *accs[mi][ni];
            const int n = n0 + ni * 16 + lr;
#pragma unroll
            for (int r = 0; r < 8; ++r) {
                int row = m0 + wv * 32 + mi * 16 + half * 8 + r;
                if (MODE == 0) {
                    C[(size_t)row * 864 + n] = a[r];
                } else if (MODE == 1) {
                    float g  = a[r] + bias[n];
                    float kv = ykv[(size_t)row * 864 + 96 + n];   // k slice
                    float vv = ykv[(size_t)row * 864 + 480 + n];  // v slice
                    C[(size_t)row * DIMC + n] = g * kv * vv;
                } else {
                    C[(size_t)row * DIMC + n] = a[r] + bias[n];
                }
            }
        }
    }
}

// ---------------------------------------------------------------------------
// Depthwise 3x3 dilated conv + channel LayerNorm + border-zeroed scatter
// into query_all. One 96-thread block (3 waves) per pixel.
// ---------------------------------------------------------------------------
__global__ __launch_bounds__(96) void conv_ln_kernel(
    const float* __restrict__ qin, int qstride,
    float* __restrict__ qout,
    float* __restrict__ qall,
    const float* __restrict__ cw,   // 96*9 (this level)
    const float* __restrict__ cb,
    const float* __restrict__ lw,
    const float* __restrict__ lb,
    int dil, int level, int start)
{
    const int pix  = blockIdx.x;
    const int c    = threadIdx.x;
    const int bimg = pix >> 12;
    const int hw   = pix & 4095;
    const int hy   = hw >> 6;
    const int wx   = hw & 63;

    float acc = cb[c];
#pragma unroll
    for (int ky = 0; ky < 3; ++ky) {
        int yy = hy + (ky - 1) * dil;
        if ((unsigned)yy < 64u) {
#pragma unroll
            for (int kx = 0; kx < 3; ++kx) {
                int xx = wx + (kx - 1) * dil;
                if ((unsigned)xx < 64u) {
                    acc += cw[c * 9 + ky * 3 + kx] *
                           qin[(size_t)((bimg << 12) + (yy << 6) + xx) * qstride + c];
                }
            }
        }
    }

    // reduce sum / sumsq across 96 channels (3 waves of 32)
    float s = acc, s2 = acc * acc;
#pragma unroll
    for (int off = 16; off > 0; off >>= 1) {
        s  += __shfl_down(s,  off, 32);
        s2 += __shfl_down(s2, off, 32);
    }
    __shared__ float ps[3], ps2[3];
    if ((c & 31) == 0) { ps[c >> 5] = s; ps2[c >> 5] = s2; }
    __syncthreads();
    float sum  = ps[0] + ps[1] + ps[2];
    float sum2 = ps2[0] + ps2[1] + ps2[2];

    const float inv = 1.0f / 96.0f;
    float mean = sum * inv;
    float var  = sum2 * inv - mean * mean;
    float o = lw[c] * ((acc - mean) * rsqrtf(var + 1e-6f)) + lb[c];

    qout[(size_t)pix * DPLC + c] = o;
    bool inside = (hy >= start) && (hy < 64 - start) &&
                  (wx >= start) && (wx < 64 - start);
    qall[(size_t)pix * DIMC + level * DPLC + c] = inside ? o : 0.0f;
}

// ---------------------------------------------------------------------------
extern "C" void kernel_launch(void* const* d_in, const int* in_sizes, int n_in,
                              void* d_out, int out_size, void* d_ws, size_t ws_size,
                              hipStream_t stream)
{
    const float* x          = (const float*)d_in[0];  // 32768 x 384
    const float* linear_w   = (const float*)d_in[1];  // 384 x 864
    const float* interact_w = (const float*)d_in[2];  // 384 x 384
    const float* interact_b = (const float*)d_in[3];  // 384
    const float* conv_ws    = (const float*)d_in[4];  // 4 x 96 x 9
    const float* conv_bs    = (const float*)d_in[5];  // 4 x 96
    const float* ln_ws      = (const float*)d_in[6];  // 4 x 96
    const float* ln_bs      = (const float*)d_in[7];  // 4 x 96
    const float* proj_w     = (const float*)d_in[8];  // 384 x 384
    const float* proj_b     = (const float*)d_in[9];  // 384

    float* ws = (float*)d_ws;
    float* y  = ws;                                   // NPIX x 864
    float* qA = y  + (size_t)NPIX * 864;              // NPIX x 96
    float* qB = qA + (size_t)NPIX * DPLC;             // NPIX x 96
    float* qa = qB + (size_t)NPIX * DPLC;             // NPIX x 384 (query_all)
    float* t  = qa + (size_t)NPIX * DIMC;             // NPIX x 384

    // 1) y = x @ linear_w
    dim3 g0(864 / 32, NPIX / 128);
    wmma_gemm_kernel<0><<<g0, 128, 0, stream>>>(x, DIMC, linear_w,
                                                nullptr, nullptr, y, 864);

    // 2) four conv+LN levels (q0 lives in y columns 0..95, stride 864)
    const float* qin_p[4]  = { y,  qA, qB, qA };
    const int    qin_s[4]  = { 864, DPLC, DPLC, DPLC };
    float*       qout_p[4] = { qA, qB, qA, qB };
    const int    dils[4]   = { 2, 3, 4, 5 };
    const int    starts[4] = { 8, 4, 2, 0 };
    for (int lv = 0; lv < 4; ++lv) {
        conv_ln_kernel<<<NPIX, 96, 0, stream>>>(
            qin_p[lv], qin_s[lv], qout_p[lv], qa,
            conv_ws + (size_t)lv * DPLC * 9, conv_bs + lv * DPLC,
            ln_ws + lv * DPLC, ln_bs + lv * DPLC,
            dils[lv], lv, starts[lv]);
    }

    // 3) t = (query_all @ interact_w^T + b) * k * v
    dim3 g1(DIMC / 32, NPIX / 128);
    wmma_gemm_kernel<1><<<g1, 128, 0, stream>>>(qa, DIMC, interact_w,
                                                interact_b, y, t, DIMC);

    // 4) out = t @ proj_w^T + proj_b
    wmma_gemm_kernel<2><<<g1, 128, 0, stream>>>(t, DIMC, proj_w,
                                                proj_b, nullptr,
                                                (float*)d_out, DIMC);
}